// KalmanFilterPredictor_47321949667423
// MI455X (gfx1250) — compile-verified
//
#include <hip/hip_runtime.h>
#include <hip/hip_bf16.h>

typedef __attribute__((ext_vector_type(2))) float v2f;
typedef __attribute__((ext_vector_type(8))) float v8f;

#define SD 14          // state dim
#define MD 7           // measurement / output dim
#define TT 512         // sequence length
#define KTOT (TT*MD)   // 3584 reduction length
#define NCHUNK (KTOT/4) // 896 WMMA k-chunks

// Workspace layout (floats)
#define OFF_A   0                         // A_t: 512 * 196
#define OFF_K   (OFF_A + TT*SD*SD)        // K_t: 512 * 98
#define OFF_M   (OFF_K + TT*SD*MD)        // M_t: 512 * 49
#define OFF_B   (OFF_M + TT*MD*MD)        // swizzled B operand: 896 * 64

// ---------------------------------------------------------------------------
// Kernel 1: batch-independent Kalman precompute (single workgroup).
// Phase A: forward Riccati recursion -> A_t, K_t
// Phase B: backward accumulation     -> M_t (7x7 per step, init folded into M_0)
// Phase C: swizzle M into WMMA_F32_16X16X4_F32 B-operand lane layout
// ---------------------------------------------------------------------------
__global__ void __launch_bounds__(256)
kalman_precompute(const float* __restrict__ Wl, const float* __restrict__ F,
                  const float* __restrict__ H,  const float* __restrict__ Q,
                  const float* __restrict__ R,  float* __restrict__ ws)
{
    __shared__ float sF[SD*SD], sQ[SD*SD], sH[MD*SD], sW[MD*SD], sR[MD*MD];
    __shared__ float P[SD*SD], Pp[SD*SD], PHt[SD*MD], Kg[SD*MD];
    __shared__ float IKH[SD*SD], Tmp[SD*SD];
    __shared__ float aug[MD][2*MD];
    __shared__ float pivinv, facs[MD], prowv[2*MD];
    __shared__ float Prow[MD*SD], Prow2[MD*SD];

    float* Abuf = ws + OFF_A;
    float* Kbuf = ws + OFF_K;
    float* Mbuf = ws + OFF_M;
    float* Bbuf = ws + OFF_B;

    const int tid = threadIdx.x;

    for (int i = tid; i < SD*SD; i += blockDim.x) { sF[i] = F[i]; sQ[i] = Q[i]; }
    for (int i = tid; i < MD*SD; i += blockDim.x) { sH[i] = H[i]; sW[i] = Wl[i]; }
    for (int i = tid; i < MD*MD; i += blockDim.x) { sR[i] = R[i]; }
    for (int i = tid; i < SD*SD; i += blockDim.x) P[i] = (i / SD == i % SD) ? 1.0f : 0.0f;
    __syncthreads();

    // ---------------- Phase A: forward Riccati ----------------
    for (int t = 0; t < TT; ++t) {
        // Tmp = P * F^T
        if (tid < SD*SD) {
            int i = tid / SD, j = tid % SD;
            float s = 0.f;
            for (int k = 0; k < SD; ++k) s += P[i*SD+k] * sF[j*SD+k];
            Tmp[tid] = s;
        }
        __syncthreads();
        // Pp = F * Tmp + Q
        if (tid < SD*SD) {
            int i = tid / SD, j = tid % SD;
            float s = sQ[tid];
            for (int k = 0; k < SD; ++k) s += sF[i*SD+k] * Tmp[k*SD+j];
            Pp[tid] = s;
        }
        __syncthreads();
        // PHt = Pp * H^T   [14x7]
        if (tid < SD*MD) {
            int i = tid / MD, j = tid % MD;
            float s = 0.f;
            for (int k = 0; k < SD; ++k) s += Pp[i*SD+k] * sH[j*SD+k];
            PHt[tid] = s;
        }
        __syncthreads();
        // aug = [ H*PHt + R | I ]   [7x14]
        if (tid < MD*MD) {
            int i = tid / MD, j = tid % MD;
            float s = sR[tid];
            for (int k = 0; k < SD; ++k) s += sH[i*SD+k] * PHt[k*MD+j];
            aug[i][j] = s;
            aug[i][j+MD] = (i == j) ? 1.f : 0.f;
        }
        __syncthreads();
        // Gauss-Jordan invert the SPD 7x7 (no pivoting needed: diag >= R)
        for (int p = 0; p < MD; ++p) {
            if (tid == 0) pivinv = 1.0f / aug[p][p];
            __syncthreads();
            if (tid < 2*MD) prowv[tid] = aug[p][tid] * pivinv;
            if (tid >= 2*MD && tid < 3*MD) facs[tid-2*MD] = aug[tid-2*MD][p];
            __syncthreads();
            if (tid < MD*2*MD) {
                int i = tid / (2*MD), j = tid % (2*MD);
                aug[i][j] = (i == p) ? prowv[j] : aug[i][j] - facs[i]*prowv[j];
            }
            __syncthreads();
        }
        // K = PHt * Sinv
        if (tid < SD*MD) {
            int i = tid / MD, j = tid % MD;
            float s = 0.f;
            for (int k = 0; k < MD; ++k) s += PHt[i*MD+k] * aug[k][MD+j];
            Kg[tid] = s;
        }
        __syncthreads();
        // IKH = I - K*H
        if (tid < SD*SD) {
            int i = tid / SD, j = tid % SD;
            float s = (i == j) ? 1.f : 0.f;
            for (int k = 0; k < MD; ++k) s -= Kg[i*MD+k] * sH[k*SD+j];
            IKH[tid] = s;
        }
        __syncthreads();
        // A_t = IKH * F ; P = IKH * Pp ; store A_t, K_t
        if (tid < SD*SD) {
            int i = tid / SD, j = tid % SD;
            float sa = 0.f, sp = 0.f;
            for (int k = 0; k < SD; ++k) {
                sa += IKH[i*SD+k] * sF[k*SD+j];
                sp += IKH[i*SD+k] * Pp[k*SD+j];
            }
            P[tid] = sp;
            Abuf[t*SD*SD + tid] = sa;
        }
        if (tid < SD*MD) Kbuf[t*SD*MD + tid] = Kg[tid];
        __syncthreads();
    }

    // ---------------- Phase B: backward pass for M_t ----------------
    // Prow = W * F  [7x14]
    if (tid < MD*SD) {
        int i = tid / SD, j = tid % SD;
        float s = 0.f;
        for (int k = 0; k < SD; ++k) s += sW[i*SD+k] * sF[k*SD+j];
        Prow[tid] = s;
    }
    __syncthreads();
    for (int t = TT - 1; t >= 0; --t) {
        const float* At = Abuf + t*SD*SD;
        const float* Kt = Kbuf + t*SD*MD;
        if (tid < MD*MD) {                       // M_t = Prow * K_t
            int i = tid / MD, j = tid % MD;
            float s = 0.f;
            for (int k = 0; k < SD; ++k) s += Prow[i*SD+k] * Kt[k*MD+j];
            Mbuf[t*MD*MD + tid] = s;
        }
        if (tid < MD*SD) {                       // Prow' = Prow * A_t
            int i = tid / SD, j = tid % SD;
            float s = 0.f;
            for (int k = 0; k < SD; ++k) s += Prow[i*SD+k] * At[k*SD+j];
            Prow2[tid] = s;
        }
        __syncthreads();
        if (tid < MD*SD) Prow[tid] = Prow2[tid];
        __syncthreads();
    }
    // fold initial-state contribution (s0 = [x0;0]) into M_0:
    // M_0[i][j] += (Prow * E)[i][j] = Prow[i][j] for j < MD
    if (tid < MD*MD) {
        int i = tid / MD, j = tid % MD;
        Mbuf[tid] += Prow[i*SD + j];
    }
    __syncthreads();

    // ---------------- Phase C: swizzle into WMMA B-operand layout ----------------
    // B matrix logical shape [3584 x 16]: row kk = t*7+j, col n = output dim (n>=7 -> 0)
    // v_wmma_f32_16x16x4_f32 B layout: lanes 0-15 hold K=4c+{0,1} (V0,V1), n=lane;
    //                                  lanes 16-31 hold K=4c+{2,3}, n=lane-16.
    for (int idx = tid; idx < NCHUNK*64; idx += blockDim.x) {
        int c    = idx >> 6;
        int lane = (idx >> 1) & 31;
        int r    = idx & 1;
        int kk   = 4*c + ((lane >> 4) << 1) + r;
        int n    = lane & 15;
        int ts   = kk / MD, j = kk % MD;
        Bbuf[idx] = (n < MD) ? Mbuf[ts*MD*MD + n*MD + j] : 0.0f;
    }
}

// ---------------------------------------------------------------------------
// Kernel 2: out[8192x7] = x_flat[8192x3584] * Mb[3584x16(:7)] + bias
// One wave per 16-row tile; 896 chained V_WMMA_F32_16X16X4_F32.
// ---------------------------------------------------------------------------
__global__ void __launch_bounds__(256)
kalman_gemm(const float* __restrict__ x, const float* __restrict__ Bbuf,
            const float* __restrict__ bias, float* __restrict__ out)
{
    const int lane = threadIdx.x & 31;
    const int wave = threadIdx.x >> 5;
    const int tile = blockIdx.x * (blockDim.x >> 5) + wave;   // 0..511
    const int row_base = tile * 16;

    // A-operand: lane<16 -> row=lane, K offsets {0,1}; lane>=16 -> row=lane-16, K {2,3}
    const int arow = row_base + (lane & 15);
    const int kofs = (lane >> 4) * 2;
    const float* __restrict__ xrow = x + (size_t)arow * KTOT + kofs;
    const float* __restrict__ brow = Bbuf + lane * 2;

    v8f c = {};
    #pragma unroll 4
    for (int ck = 0; ck < NCHUNK; ++ck) {
        v2f a = *(const v2f*)(xrow + ck * 4);
        v2f b = *(const v2f*)(brow + ck * 64);
        c = __builtin_amdgcn_wmma_f32_16x16x4_f32(
                /*neg_a=*/false, a, /*neg_b=*/false, b,
                /*c_mod=*/(short)0, c, /*reuse_a=*/false, /*reuse_b=*/false);
    }

    // C/D layout: VGPR r holds row M=r (lanes 0-15) / M=r+8 (lanes 16-31), N=lane&15
    const int n = lane & 15;
    const int rofs = (lane >> 4) * 8;
    if (n < MD) {
        const float bn = bias[n];
        #pragma unroll
        for (int r = 0; r < 8; ++r) {
            out[(size_t)(row_base + rofs + r) * MD + n] = c[r] + bn;
        }
    }
}

extern "C" void kernel_launch(void* const* d_in, const int* in_sizes, int n_in,
                              void* d_out, int out_size, void* d_ws, size_t ws_size,
                              hipStream_t stream) {
    const float* x  = (const float*)d_in[0];  // [8192,512,7]
    const float* W  = (const float*)d_in[1];  // [7,14]
    const float* b  = (const float*)d_in[2];  // [7]
    const float* F  = (const float*)d_in[3];  // [14,14]
    const float* H  = (const float*)d_in[4];  // [7,14]
    const float* Q  = (const float*)d_in[5];  // [14,14]
    const float* R  = (const float*)d_in[6];  // [7,7]
    float* out = (float*)d_out;
    float* ws  = (float*)d_ws;

    kalman_precompute<<<1, 256, 0, stream>>>(W, F, H, Q, R, ws);

    // 8192/16 = 512 tiles, 8 waves (256 threads) per block -> 64 blocks
    kalman_gemm<<<64, 256, 0, stream>>>(x, ws + OFF_B, b, out);
}